// DGLMaxPoolAggregator_4733053960602
// MI455X (gfx1250) — compile-verified
//
#include <hip/hip_runtime.h>

typedef __attribute__((ext_vector_type(2))) float v2f;
typedef __attribute__((ext_vector_type(8))) float v8f;

#define N_NODES 100000
#define N_EDGES 1600000
#define D 128
#define NEG_INF (-3.402823466e+38f)

// ---------------------------------------------------------------------------
// K1: copy feat -> out[:, 0:128] (float4 per lane), zero degree histogram.
// grid: N_NODES*32/256 = 12500 blocks x 256
// ---------------------------------------------------------------------------
__global__ void copy_init_kernel(const float* __restrict__ feat,
                                 float* __restrict__ out,
                                 int* __restrict__ deg) {
    int idx = blockIdx.x * blockDim.x + threadIdx.x;   // 0 .. N*32-1
    int n  = idx >> 5;
    int j4 = (idx & 31) << 2;
    float4 v = *(const float4*)(feat + (size_t)n * D + j4);
    *(float4*)(out + (size_t)n * (2 * D) + j4) = v;
    if (idx <= N_NODES) deg[idx] = 0;                  // N+1 entries
}

// ---------------------------------------------------------------------------
// K2: norm_h = feat @ W1^T via V_WMMA_F32_16X16X4_F32.
// One wave per 32-row stripe: 2 A tiles x 8 B tiles = 16 accumulators
// (128 VGPRs) + b[8] staging (16 VGPRs). __launch_bounds__(160,1) raises
// the VGPR budget so nothing spills; occupancy of 5 waves/WGP is fine for
// a streaming GEMM. Per k-step: 2 A + 8 B loads (one clause, staggered
// waits) feed 16 WMMAs; each B register is consumed by two WMMAs.
// feat streamed exactly once; W1 (64KB) stays hot in WGP$/L2.
// Stripes: 100000/32 = 3125; 5 waves/block -> 625 blocks x 160 threads.
// A layout (16x4 f32): lanes 0-15 hold K={0,1}, lanes 16-31 hold K={2,3}.
// B[k][o] = W1[o][k]  -> identical load pattern with W1 as base.
// C layout: VGPR v -> (M = v + 8*half, N = lane&15).
// ---------------------------------------------------------------------------
__global__ void __launch_bounds__(160, 1)
gemm_wmma_kernel(const float* __restrict__ feat,
                 const float* __restrict__ W1,
                 float* __restrict__ norm_h) {
    int wave = threadIdx.x >> 5;
    int lane = threadIdx.x & 31;
    int tm = blockIdx.x * 5 + wave;      // 32-row stripe, 0..3124
    int half = lane >> 4;                // 0|1
    int l15  = lane & 15;

    const float* arow0 = feat + (size_t)(tm * 32 + l15) * D + half * 2;
    const float* arow1 = arow0 + (size_t)16 * D;
    const float* bbase = W1   + (size_t)l15 * D + half * 2;

    v8f c0[8] = {};
    v8f c1[8] = {};
    for (int k = 0; k < D; k += 4) {
        v2f a0 = *(const v2f*)(arow0 + k);
        v2f a1 = *(const v2f*)(arow1 + k);
        v2f b[8];
#pragma unroll
        for (int tn = 0; tn < 8; ++tn) {
            b[tn] = *(const v2f*)(bbase + (size_t)tn * 16 * D + k);
        }
#pragma unroll
        for (int tn = 0; tn < 8; ++tn) {
            c0[tn] = __builtin_amdgcn_wmma_f32_16x16x4_f32(
                         false, a0, false, b[tn], (short)0, c0[tn], false, false);
            c1[tn] = __builtin_amdgcn_wmma_f32_16x16x4_f32(
                         false, a1, false, b[tn], (short)0, c1[tn], false, false);
        }
    }

    float* crow0 = norm_h + (size_t)(tm * 32 + half * 8) * D + l15;
    float* crow1 = crow0 + (size_t)16 * D;
#pragma unroll
    for (int tn = 0; tn < 8; ++tn) {
#pragma unroll
        for (int v = 0; v < 8; ++v) {
            crow0[(size_t)v * D + tn * 16] = c0[tn][v];
            crow1[(size_t)v * D + tn * 16] = c1[tn][v];
        }
    }
}

// ---------------------------------------------------------------------------
// K3: degree histogram over dst.  grid: E/256 = 6250 blocks x 256
// ---------------------------------------------------------------------------
__global__ void hist_kernel(const int* __restrict__ dst, int* __restrict__ deg) {
    int e = blockIdx.x * blockDim.x + threadIdx.x;
    atomicAdd(&deg[dst[e]], 1);
}

// ---------------------------------------------------------------------------
// K4: exclusive scan of deg[0..N) -> offsets[0..N], cursor[0..N).
// Single block, 1024 threads, Hillis-Steele in LDS with running carry.
// ---------------------------------------------------------------------------
__global__ void scan_kernel(const int* __restrict__ deg,
                            int* __restrict__ offsets,
                            int* __restrict__ cursor) {
    __shared__ int sdata[1024];
    __shared__ int carry;
    int tid = threadIdx.x;
    if (tid == 0) carry = 0;
    __syncthreads();
    for (int base = 0; base <= N_NODES; base += 1024) {
        int i = base + tid;
        int x = (i < N_NODES) ? deg[i] : 0;
        sdata[tid] = x;
        __syncthreads();
#pragma unroll
        for (int off = 1; off < 1024; off <<= 1) {
            int t = (tid >= off) ? sdata[tid - off] : 0;
            __syncthreads();
            sdata[tid] += t;
            __syncthreads();
        }
        int excl = carry + sdata[tid] - x;
        if (i <= N_NODES) {
            offsets[i] = excl;
            if (i < N_NODES) cursor[i] = excl;
        }
        __syncthreads();
        if (tid == 1023) carry += sdata[1023];
        __syncthreads();
    }
}

// ---------------------------------------------------------------------------
// K5: scatter edges into CSR slots.  grid: 6250 x 256
// ---------------------------------------------------------------------------
__global__ void scatter_kernel(const int* __restrict__ src,
                               const int* __restrict__ dst,
                               int* __restrict__ cursor,
                               int* __restrict__ csr_src) {
    int e = blockIdx.x * blockDim.x + threadIdx.x;
    int pos = atomicAdd(&cursor[dst[e]], 1);
    csr_src[pos] = src[e];
}

// ---------------------------------------------------------------------------
// K6: per-node gather max. One wave per node, lane owns 4 features (float4).
// norm_h (51MB) fits in 192MB L2 -> gather hits L2.
// grid: N/8 = 12500 blocks x 256 (8 waves/block)
// ---------------------------------------------------------------------------
__global__ void segmax_kernel(const float* __restrict__ norm_h,
                              const int* __restrict__ offsets,
                              const int* __restrict__ csr_src,
                              float* __restrict__ out) {
    int wave = threadIdx.x >> 5;
    int lane = threadIdx.x & 31;
    int node = blockIdx.x * 8 + wave;   // < N_NODES exactly (12500*8)
    int beg = offsets[node];
    int end = offsets[node + 1];
    float4 acc = make_float4(NEG_INF, NEG_INF, NEG_INF, NEG_INF);
    int j4 = lane << 2;
    for (int e = beg; e < end; ++e) {
        int s = csr_src[e];
        float4 v = *(const float4*)(norm_h + (size_t)s * D + j4);
        acc.x = fmaxf(acc.x, v.x);
        acc.y = fmaxf(acc.y, v.y);
        acc.z = fmaxf(acc.z, v.z);
        acc.w = fmaxf(acc.w, v.w);
    }
    *(float4*)(out + (size_t)node * (2 * D) + D + j4) = acc;
}

// ---------------------------------------------------------------------------
extern "C" void kernel_launch(void* const* d_in, const int* in_sizes, int n_in,
                              void* d_out, int out_size, void* d_ws, size_t ws_size,
                              hipStream_t stream) {
    const float* feat = (const float*)d_in[0];
    const float* W1   = (const float*)d_in[1];
    const int*   src  = (const int*)d_in[2];
    const int*   dst  = (const int*)d_in[3];
    float* out = (float*)d_out;

    // workspace carve (16B aligned sections)
    char* ws = (char*)d_ws;
    const size_t NORM_H_BYTES = (size_t)N_NODES * D * sizeof(float); // 51,200,000
    const size_t INTARR_BYTES = 400064;                              // (N+1) ints padded
    float* norm_h  = (float*)ws;
    int*   deg     = (int*)(ws + NORM_H_BYTES);
    int*   offsets = (int*)(ws + NORM_H_BYTES + INTARR_BYTES);
    int*   cursor  = (int*)(ws + NORM_H_BYTES + 2 * INTARR_BYTES);
    int*   csr_src = (int*)(ws + NORM_H_BYTES + 3 * INTARR_BYTES);

    copy_init_kernel<<<N_NODES * 32 / 256, 256, 0, stream>>>(feat, out, deg);
    gemm_wmma_kernel<<<N_NODES / 32 / 5, 160, 0, stream>>>(feat, W1, norm_h);
    hist_kernel<<<N_EDGES / 256, 256, 0, stream>>>(dst, deg);
    scan_kernel<<<1, 1024, 0, stream>>>(deg, offsets, cursor);
    scatter_kernel<<<N_EDGES / 256, 256, 0, stream>>>(src, dst, cursor, csr_src);
    segmax_kernel<<<N_NODES / 8, 256, 0, stream>>>(norm_h, offsets, csr_src, out);
}